// MulticoreBPFLayer_19258633355310
// MI455X (gfx1250) — compile-verified
//
#include <hip/hip_runtime.h>

typedef float v2f __attribute__((ext_vector_type(2)));
typedef float v8f __attribute__((ext_vector_type(8)));

// Streaming pass: per 16-particle tile per wave,
//   y = T*x + noise                   (VALU, 9 FMA)
//   pred(16x64) = y * F^T             (4x v_wmma_f32_16x16x4_f32, C = inline 0)
//   w = rowsum((pred - inputs)^2)     (7-shuffle butterfly transpose-reduce)
//   accumulate Sum(w), Sum(w*y)       (weighted mean == E[categorical resample mean])
__global__ __launch_bounds__(256) void bpf_pass(
    const float* __restrict__ inp,    // 64
    const float* __restrict__ state,  // N x 3
    const float* __restrict__ Tm,     // 3 x 3
    const float* __restrict__ Fm,     // 64 x 3
    const float* __restrict__ noise,  // N x 3
    float* __restrict__ acc4,         // [W, S0, S1, S2]
    int N)
{
  const int lane = threadIdx.x & 31;
  const int wib  = threadIdx.x >> 5;
  const int wpb  = blockDim.x >> 5;
  const int wid  = blockIdx.x * wpb + wib;
  const int nw   = gridDim.x * wpb;

  // Uniform 3x3 transition matrix.
  const float t00 = Tm[0], t01 = Tm[1], t02 = Tm[2];
  const float t10 = Tm[3], t11 = Tm[4], t12 = Tm[5];
  const float t20 = Tm[6], t21 = Tm[7], t22 = Tm[8];

  const int  col = lane & 15;   // particle-in-tile (A rows / D columns index)
  const bool hiK = lane >= 16;  // lanes 16-31 carry K=2,3 for A and B

  // Loop-invariant B tiles (4x16 = F^T columns j*16..j*16+15, K padded to 4)
  // and the per-lane channel offset c_j = -inputs[n] folded into the squaring.
  v2f   Bt[4];
  float Cc[4];
#pragma unroll
  for (int j = 0; j < 4; ++j) {
    const int n  = j * 16 + col;
    const float f0 = Fm[n * 3 + 0];
    const float f1 = Fm[n * 3 + 1];
    const float f2 = Fm[n * 3 + 2];
    Bt[j].x = hiK ? f2 : f0;     // B[K=0 or 2][n]
    Bt[j].y = hiK ? 0.0f : f1;   // B[K=1 or 3][n], K=3 is pad
    Cc[j]   = -inp[n];
  }

  const bool b0 = (lane & 1) != 0;
  const bool b1 = (lane & 2) != 0;
  const bool b2 = (lane & 4) != 0;
  const bool midOct = (((lane >> 3) ^ (lane >> 4)) & 1) != 0;

  double aW = 0.0, aS0 = 0.0, aS1 = 0.0, aS2 = 0.0;

  const int numTiles = (N + 15) >> 4;
  for (int tile = wid; tile < numTiles; tile += nw) {   // wave-uniform loop
    const int  p     = tile * 16 + col;
    const bool valid = (p < N);
    const int  pc    = valid ? p : (N - 1);

    const float x0 = state[pc * 3 + 0];
    const float x1 = state[pc * 3 + 1];
    const float x2 = state[pc * 3 + 2];
    const float n0 = noise[pc * 3 + 0];
    const float n1 = noise[pc * 3 + 1];
    const float n2 = noise[pc * 3 + 2];

    // new_state: y = T @ x + noise
    const float y0 = fmaf(t00, x0, fmaf(t01, x1, fmaf(t02, x2, n0)));
    const float y1 = fmaf(t10, x0, fmaf(t11, x1, fmaf(t12, x2, n1)));
    const float y2 = fmaf(t20, x0, fmaf(t21, x1, fmaf(t22, x2, n2)));

    // A-matrix 16x4 f32: lanes 0-15 -> (K0,K1), lanes 16-31 -> (K2,K3=0)
    v2f A;
    A.x = hiK ? y2 : y0;
    A.y = hiK ? 0.0f : y1;

    // D_j = A*B_j + 0 (16x16 predicted tile); square (pred - inputs).
    float sq[8] = {0, 0, 0, 0, 0, 0, 0, 0};
#pragma unroll
    for (int j = 0; j < 4; ++j) {
      v8f Zc = {};  // inline-0 C operand
      v8f D = __builtin_amdgcn_wmma_f32_16x16x4_f32(
          false, A, false, Bt[j], (short)0, Zc, false, false);
      const float c = Cc[j];
#pragma unroll
      for (int v = 0; v < 8; ++v) {
        const float r = D[v] + c;
        sq[v] = fmaf(r, r, sq[v]);
      }
    }

    // Butterfly transpose-reduce over the 8-lane column group:
    // 7 shuffles leave t = sum over group columns for particle row v = lane&7.
    float p4[4];
#pragma unroll
    for (int i = 0; i < 4; ++i) {
      const float give = b0 ? sq[2 * i] : sq[2 * i + 1];
      const float keep = b0 ? sq[2 * i + 1] : sq[2 * i];
      p4[i] = keep + __shfl_xor(give, 1, 32);
    }
    float p2[2];
#pragma unroll
    for (int i = 0; i < 2; ++i) {
      const float give = b1 ? p4[2 * i] : p4[2 * i + 1];
      const float keep = b1 ? p4[2 * i + 1] : p4[2 * i];
      p2[i] = keep + __shfl_xor(give, 2, 32);
    }
    float t;
    {
      const float give = b2 ? p2[0] : p2[1];
      const float keep = b2 ? p2[1] : p2[0];
      t = keep + __shfl_xor(give, 4, 32);
    }
    // Combine the two column groups, then swap middle octets so that
    // lane L holds w[particle lane&15] (matching its y0..y2).
    t += __shfl_xor(t, 8, 32);
    const float o = __shfl_xor(t, 24, 32);
    float w = midOct ? o : t;
    if (!valid) w = 0.0f;

    // Each particle is counted by 2 lanes; the factor cancels in S/W.
    aW  += (double)w;
    aS0 += (double)(w * y0);
    aS1 += (double)(w * y1);
    aS2 += (double)(w * y2);
  }

  // Wave reduction in f32 (per-lane partials were kept in f64), 1 atomic set per wave.
  float rW = (float)aW, r0 = (float)aS0, r1 = (float)aS1, r2 = (float)aS2;
#pragma unroll
  for (int m = 1; m < 32; m <<= 1) {
    rW += __shfl_xor(rW, m, 32);
    r0 += __shfl_xor(r0, m, 32);
    r1 += __shfl_xor(r1, m, 32);
    r2 += __shfl_xor(r2, m, 32);
  }
  if (lane == 0) {
    atomicAdd(&acc4[0], rW);
    atomicAdd(&acc4[1], r0);
    atomicAdd(&acc4[2], r1);
    atomicAdd(&acc4[3], r2);
  }
}

__global__ void bpf_finalize(const float* __restrict__ acc4, float* __restrict__ out) {
  const int d = threadIdx.x;
  if (d < 3) out[d] = acc4[1 + d] / acc4[0];
}

extern "C" void kernel_launch(void* const* d_in, const int* in_sizes, int n_in,
                              void* d_out, int out_size, void* d_ws, size_t ws_size,
                              hipStream_t stream) {
  const float* inputs = (const float*)d_in[0];  // (64,)
  const float* state  = (const float*)d_in[1];  // (N,3)
  const float* Tm     = (const float*)d_in[2];  // (3,3)
  const float* Fm     = (const float*)d_in[3];  // (64,3)
  const float* noise  = (const float*)d_in[4];  // (N,3)
  float* out  = (float*)d_out;                  // (3,)
  float* acc4 = (float*)d_ws;                   // [W,S0,S1,S2]

  const int N = in_sizes[1] / 3;

  hipMemsetAsync(acc4, 0, 4 * sizeof(float), stream);
  bpf_pass<<<1024, 256, 0, stream>>>(inputs, state, Tm, Fm, noise, acc4, N);
  bpf_finalize<<<1, 32, 0, stream>>>(acc4, out);
}